// GCN_13494787244545
// MI455X (gfx1250) — compile-verified
//
#include <hip/hip_runtime.h>

typedef float v2f __attribute__((ext_vector_type(2)));
typedef float v8f __attribute__((ext_vector_type(8)));

#define F_DIM 128
#define U_DIM 128

// ---------- degree: deg[i] starts at 1.0 (self-loop), += edge weights on row ----------
__global__ void k_init_deg(float* __restrict__ deg, int n) {
  int i = blockIdx.x * blockDim.x + threadIdx.x;
  if (i < n) deg[i] = 1.0f;
}

__global__ void k_accum_deg(const int* __restrict__ row, const float* __restrict__ ew,
                            float* __restrict__ deg, int e) {
  int i = blockIdx.x * blockDim.x + threadIdx.x;
  if (i < e) atomicAdd(&deg[row[i]], ew[i]);
}

__global__ void k_dinv(float* __restrict__ deg, int n) {
  int i = blockIdx.x * blockDim.x + threadIdx.x;
  if (i < n) {
    float d = deg[i];
    deg[i] = (d > 0.0f) ? rsqrtf(d) : 0.0f;   // deg >= 1 in practice
  }
}

// ---------- dense transform h = x @ W via V_WMMA_F32_16X16X4_F32 ----------
// 256 threads = 8 waves; wave w computes the 16x16 tile at cols [16w,16w+16),
// block handles rows [16*blockIdx.x, +16). K loop: 128/4 = 32 WMMAs per wave.
__global__ void k_gemm_wmma(const float* __restrict__ x, const float* __restrict__ w,
                            float* __restrict__ h, int n) {
  const int lane    = threadIdx.x & 31;
  const int wv      = threadIdx.x >> 5;          // 0..7: column tile
  const int rowBase = blockIdx.x * 16;
  const int colBase = wv * 16;
  const int nlane   = lane & 15;                 // N (and M for A loads)
  const int khalf   = (lane >> 4) << 1;          // 0 or 2 (ISA A-layout 16x4 f32)

  int mrow = rowBase + nlane;
  if (mrow >= n) mrow = n - 1;                   // clamp loads; stores guarded below
  const float* arow = x + (size_t)mrow * F_DIM;
  const float* bcol = w + colBase + nlane;

  v8f acc = {};
#pragma unroll
  for (int k = 0; k < F_DIM; k += 4) {
    // A 16x4: lane<16 -> (K=k, k+1); lane>=16 -> (K=k+2, k+3)
    v2f a = *(const v2f*)(arow + k + khalf);
    // B 4x16 mirrored: VGPR0 = row k+khalf, VGPR1 = row k+khalf+1, lane = N
    v2f b;
    b.x = bcol[(size_t)(k + khalf) * U_DIM];
    b.y = bcol[(size_t)(k + khalf + 1) * U_DIM];
    acc = __builtin_amdgcn_wmma_f32_16x16x4_f32(false, a, false, b,
                                                (short)0, acc, false, false);
  }

  // C/D layout: lane<16 -> VGPR j = (M=j, N=lane); lane>=16 -> (M=8+j, N=lane-16)
  const int mBase = rowBase + ((lane >> 4) << 3);
  float* hout = h + (size_t)colBase + nlane;
#pragma unroll
  for (int j = 0; j < 8; ++j) {
    int m = mBase + j;
    if (m < n) hout[(size_t)m * U_DIM] = acc[j];
  }
}

// ---------- out init + self-loop term: out[i] = dinv[i]^2 * h[i] ----------
__global__ void k_selfloop(const float* __restrict__ h, const float* __restrict__ dinv,
                           float* __restrict__ out, int n) {
  int t = blockIdx.x * blockDim.x + threadIdx.x;       // n*32 float4-threads
  if (t < n * 32) {
    int i = t >> 5;
    float s = dinv[i];
    s = s * s;
    float4 v = ((const float4*)h)[t];
    float4 o = make_float4(s * v.x, s * v.y, s * v.z, s * v.w);
    ((float4*)out)[t] = o;
  }
}

// ---------- edge scatter: one wave per edge, 32 lanes x float4 = 128 floats ----------
__global__ void k_scatter(const int* __restrict__ row, const int* __restrict__ col,
                          const float* __restrict__ ew, const float* __restrict__ dinv,
                          const float* __restrict__ h, float* __restrict__ out, int e) {
  long long t = (long long)blockIdx.x * blockDim.x + threadIdx.x;
  int eid  = (int)(t >> 5);
  if (eid >= e) return;
  int lane = (int)(t & 31);
  int r = row[eid];
  int c = col[eid];
  float wn = dinv[r] * ew[eid] * dinv[c];
  float4 v = ((const float4*)(h + (size_t)c * U_DIM))[lane];  // coalesced 512B row
  float* o = out + (size_t)r * U_DIM + lane * 4;
  atomicAdd(o + 0, wn * v.x);
  atomicAdd(o + 1, wn * v.y);
  atomicAdd(o + 2, wn * v.z);
  atomicAdd(o + 3, wn * v.w);
}

// ---------- bias + relu (in place on d_out) ----------
__global__ void k_bias_relu(float* __restrict__ out, const float* __restrict__ bias, int n) {
  int t = blockIdx.x * blockDim.x + threadIdx.x;
  if (t < n * 32) {
    int u4 = (t & 31) << 2;
    float4 v = ((float4*)out)[t];
    float4 b = *(const float4*)(bias + u4);
    v.x = fmaxf(v.x + b.x, 0.0f);
    v.y = fmaxf(v.y + b.y, 0.0f);
    v.z = fmaxf(v.z + b.z, 0.0f);
    v.w = fmaxf(v.w + b.w, 0.0f);
    ((float4*)out)[t] = v;
  }
}

extern "C" void kernel_launch(void* const* d_in, const int* in_sizes, int n_in,
                              void* d_out, int out_size, void* d_ws, size_t ws_size,
                              hipStream_t stream) {
  const float* x    = (const float*)d_in[0];
  const int*   ei   = (const int*)d_in[1];
  const float* ew   = (const float*)d_in[2];
  const float* krn  = (const float*)d_in[3];
  const float* bias = (const float*)d_in[4];

  const int n = in_sizes[0] / F_DIM;   // num nodes
  const int e = in_sizes[2];           // num edges
  const int* row = ei;                 // edge_index[0]
  const int* col = ei + e;             // edge_index[1]

  // workspace: [deg/dinv : n floats][h : n*U floats]  (~51.6 MB total)
  char*  ws   = (char*)d_ws;
  float* dinv = (float*)ws;
  size_t hoff = (((size_t)n * sizeof(float)) + 255) & ~(size_t)255;
  float* h    = (float*)(ws + hoff);
  float* out  = (float*)d_out;

  const int B = 256;
  k_init_deg <<<(n + B - 1) / B, B, 0, stream>>>(dinv, n);
  k_accum_deg<<<(e + B - 1) / B, B, 0, stream>>>(row, ew, dinv, e);
  k_dinv     <<<(n + B - 1) / B, B, 0, stream>>>(dinv, n);

  k_gemm_wmma<<<(n + 15) / 16, 256, 0, stream>>>(x, krn, h, n);

  int nt = n * 32;                     // float4-threads over [n,128]
  k_selfloop <<<(nt + B - 1) / B, B, 0, stream>>>(h, dinv, out, n);

  long long st = (long long)e * 32;    // one wave (32 lanes) per edge
  k_scatter  <<<(int)((st + B - 1) / B), B, 0, stream>>>(row, col, ew, dinv, h, out, e);

  k_bias_relu<<<(nt + B - 1) / B, B, 0, stream>>>(out, bias, n);
}